// SPDNetwork_1_63831803953807
// MI455X (gfx1250) — compile-verified
//
#include <hip/hip_runtime.h>

// SPDNet forward, fully fused, one wave (wave32) per batch element.
//   Stage 1: Y1 = w1^T X w1 via V_WMMA_F32_16X16X4_F32 (f32 precision)
//   Stage 2: parallel-ordered Jacobi eigh (32x32) in LDS -> U, lam
//   Stage 3: G = w2^T U (WMMA), Y2 = (G*lam) G^T (WMMA)   [ReEig folded]
//   Stage 4: Jacobi eigh (16x16) -> U16, lam2
//   Stage 5: H = w3^T U16, Y3 = H*lam2*H^T (VALU, tiny)
//   Stage 6: scalar 4x4 Jacobi log-eig + FC + log_softmax on lane 0
// HBM traffic: 128MB in + 2.4MB out => ~5.6us floor; eigensolver (LDS/VALU)
// dominates, so everything stays resident in LDS, zero workspace.

typedef __attribute__((ext_vector_type(2))) float v2f;
typedef __attribute__((ext_vector_type(8))) float v8f;

#define LDX 33              // row stride (pad) -> conflict-free column walks
#define WAVES_PER_BLOCK 4
#define REC_EPS 1e-4f
#define LOG_EPS 1e-10f

struct WaveScratch {
  float A[32 * LDX];   // matrix under diagonalization / tile scratch
  float U[32 * LDX];   // accumulated eigenvectors / GEMM temp
  float cs[32];        // per-pair (c,s)
  float lam[32];       // clamped eigenvalues
  int   pq[32];        // per-pair (p,q)
};

__device__ __forceinline__ v8f wmma4(v2f a, v2f b, v8f c) {
  // D = A(16x4) * B(4x16) + C, full f32
  return __builtin_amdgcn_wmma_f32_16x16x4_f32(false, a, false, b,
                                               (short)0, c, false, false);
}

// chess-tournament pairing: round r (0..N-2), pair k (0..N/2-1)
__device__ __forceinline__ void pair_for(int N, int r, int k, int* p, int* q) {
  int a, b;
  if (k == 0) { a = r; b = N - 1; }
  else {
    a = (r + k) % (N - 1);
    b = (r + (N - 1) - k) % (N - 1);
  }
  *p = a < b ? a : b;
  *q = a < b ? b : a;
}

// Parallel-ordered cyclic Jacobi on NxN symmetric matrix in LDS.
// A -> ~diag(eigenvalues); U -> eigenvectors (A0 = U diag U^T).
template <int N, int SWEEPS>
__device__ void jacobi_eigh(float* A, float* U, float* cs, int* pq, int lane) {
  for (int i = lane; i < N * N; i += 32) {
    int r = i / N, c = i % N;
    U[r * LDX + c] = (r == c) ? 1.0f : 0.0f;
  }
  __syncthreads();
  for (int sweep = 0; sweep < SWEEPS; ++sweep) {
    for (int r = 0; r < N - 1; ++r) {
      // --- rotation coefficients: one pair per lane (disjoint index pairs)
      if (lane < N / 2) {
        int p, q; pair_for(N, r, lane, &p, &q);
        float app = A[p * LDX + p];
        float aqq = A[q * LDX + q];
        float apq = A[p * LDX + q];
        float c = 1.0f, s = 0.0f;
        if (fabsf(apq) > 1e-30f) {
          float tau = (aqq - app) / (2.0f * apq);
          float t = copysignf(1.0f, tau) / (fabsf(tau) + sqrtf(1.0f + tau * tau));
          c = 1.0f / sqrtf(1.0f + t * t);
          s = t * c;
        }
        cs[2 * lane] = c; cs[2 * lane + 1] = s;
        pq[2 * lane] = p; pq[2 * lane + 1] = q;
      }
      __syncthreads();
      // --- A <- J^T A   (lane j updates column j; pairs are disjoint rows)
      if (lane < N) {
        int j = lane;
        #pragma unroll
        for (int k = 0; k < N / 2; ++k) {
          int p = pq[2 * k], q = pq[2 * k + 1];
          float c = cs[2 * k], s = cs[2 * k + 1];
          float x = A[p * LDX + j], y = A[q * LDX + j];
          A[p * LDX + j] = c * x - s * y;
          A[q * LDX + j] = s * x + c * y;
        }
      }
      __syncthreads();
      // --- A <- A J, U <- U J  (lane i updates row i; disjoint columns)
      if (lane < N) {
        int i = lane;
        #pragma unroll
        for (int k = 0; k < N / 2; ++k) {
          int p = pq[2 * k], q = pq[2 * k + 1];
          float c = cs[2 * k], s = cs[2 * k + 1];
          float x = A[i * LDX + p], y = A[i * LDX + q];
          A[i * LDX + p] = c * x - s * y;
          A[i * LDX + q] = s * x + c * y;
          float ux = U[i * LDX + p], uy = U[i * LDX + q];
          U[i * LDX + p] = c * ux - s * uy;
          U[i * LDX + q] = s * ux + c * uy;
        }
      }
      __syncthreads();
    }
  }
}

__global__ __launch_bounds__(WAVES_PER_BLOCK * 32)
void spdnet_fused_kernel(const float* __restrict__ x,
                         const float* __restrict__ w1,
                         const float* __restrict__ w2,
                         const float* __restrict__ w3,
                         const float* __restrict__ fcw,
                         float* __restrict__ out, int B) {
  __shared__ WaveScratch sm[WAVES_PER_BLOCK];
  const int lane = threadIdx.x & 31;
  const int wid  = threadIdx.x >> 5;
  const int gw   = blockIdx.x * WAVES_PER_BLOCK + wid;   // global wave id
  const int b    = gw < B ? gw : (B - 1);                // clamp: keep barriers uniform

  WaveScratch& W = sm[wid];
  float* As = W.A;
  float* Us = W.U;

  const float* Xb = x + (size_t)b * 1024;

  // ---- load X (32x32) into LDS
  #pragma unroll
  for (int t = 0; t < 32; ++t)
    As[t * LDX + lane] = Xb[t * 32 + lane];
  __syncthreads();

  const int ml = lane & 15;          // fragment row/col within tile
  const int kh = (lane >> 4) << 1;   // k-offset per lane half: 0 or 2

  // ---- GEMM1: T = w1^T * X  -> Us   (A-frag = w1^T, B-frag from LDS)
  #pragma unroll
  for (int ti = 0; ti < 2; ++ti)
  #pragma unroll
  for (int tj = 0; tj < 2; ++tj) {
    v8f acc = {};
    #pragma unroll
    for (int kk = 0; kk < 8; ++kk) {
      int k0 = kk * 4 + kh;
      v2f a, bf;
      a.x  = w1[(k0 + 0) * 32 + ti * 16 + ml];   // (w1^T)[m][k] = w1[k][m]
      a.y  = w1[(k0 + 1) * 32 + ti * 16 + ml];
      bf.x = As[(k0 + 0) * LDX + tj * 16 + ml];  // X[k][n]
      bf.y = As[(k0 + 1) * LDX + tj * 16 + ml];
      acc = wmma4(a, bf, acc);
    }
    #pragma unroll
    for (int v = 0; v < 8; ++v)
      Us[(ti * 16 + v + ((lane >> 4) << 3)) * LDX + tj * 16 + ml] = acc[v];
  }
  __syncthreads();

  // ---- GEMM2: Y1 = T * w1 -> As (overwrites X)
  #pragma unroll
  for (int ti = 0; ti < 2; ++ti)
  #pragma unroll
  for (int tj = 0; tj < 2; ++tj) {
    v8f acc = {};
    #pragma unroll
    for (int kk = 0; kk < 8; ++kk) {
      int k0 = kk * 4 + kh;
      v2f a, bf;
      a.x  = Us[(ti * 16 + ml) * LDX + k0 + 0];  // T[m][k]
      a.y  = Us[(ti * 16 + ml) * LDX + k0 + 1];
      bf.x = w1[(k0 + 0) * 32 + tj * 16 + ml];   // w1[k][n]
      bf.y = w1[(k0 + 1) * 32 + tj * 16 + ml];
      acc = wmma4(a, bf, acc);
    }
    #pragma unroll
    for (int v = 0; v < 8; ++v)
      As[(ti * 16 + v + ((lane >> 4) << 3)) * LDX + tj * 16 + ml] = acc[v];
  }
  __syncthreads();

  // ---- eigh 32x32 ; ReEig clamp
  jacobi_eigh<32, 10>(As, Us, W.cs, W.pq, lane);
  W.lam[lane] = fmaxf(As[lane * LDX + lane], REC_EPS);
  __syncthreads();

  // ---- G = w2^T * U32  (16x32) -> As rows 0..15
  #pragma unroll
  for (int tj = 0; tj < 2; ++tj) {
    v8f acc = {};
    #pragma unroll
    for (int kk = 0; kk < 8; ++kk) {
      int k0 = kk * 4 + kh;
      v2f a, bf;
      a.x  = w2[(k0 + 0) * 16 + ml];             // (w2^T)[m][k] = w2[k][m]
      a.y  = w2[(k0 + 1) * 16 + ml];
      bf.x = Us[(k0 + 0) * LDX + tj * 16 + ml];  // U[k][n]
      bf.y = Us[(k0 + 1) * LDX + tj * 16 + ml];
      acc = wmma4(a, bf, acc);
    }
    #pragma unroll
    for (int v = 0; v < 8; ++v)
      As[(v + ((lane >> 4) << 3)) * LDX + tj * 16 + ml] = acc[v];
  }
  __syncthreads();

  // ---- Y2 = (G * diag(lam)) * G^T  (16x16, single tile)
  {
    v8f acc = {};
    #pragma unroll
    for (int kk = 0; kk < 8; ++kk) {
      int k0 = kk * 4 + kh;
      v2f a, bf;
      bf.x = As[ml * LDX + k0 + 0];        // G^T[k][n] = G[n][k]
      bf.y = As[ml * LDX + k0 + 1];
      a.x  = bf.x * W.lam[k0 + 0];         // (G*lam)[m][k] (same address, m==n==ml)
      a.y  = bf.y * W.lam[k0 + 1];
      acc = wmma4(a, bf, acc);
    }
    __syncthreads();
    #pragma unroll
    for (int v = 0; v < 8; ++v)
      As[(v + ((lane >> 4) << 3)) * LDX + ml] = acc[v];
  }
  __syncthreads();

  // ---- eigh 16x16 ; ReEig clamp
  jacobi_eigh<16, 8>(As, Us, W.cs, W.pq, lane);
  if (lane < 16) W.lam[lane] = fmaxf(As[lane * LDX + lane], REC_EPS);
  __syncthreads();

  // ---- H = w3^T * U16 (4x16) -> As rows 16..19
  #pragma unroll
  for (int rep = 0; rep < 2; ++rep) {
    int id = lane + rep * 32;      // 0..63
    int i = id >> 4, t = id & 15;
    float s = 0.f;
    #pragma unroll
    for (int j = 0; j < 16; ++j)
      s += w3[j * 4 + i] * Us[j * LDX + t];
    As[(16 + i) * LDX + t] = s;
  }
  __syncthreads();

  // ---- Y3 = H * diag(lam2) * H^T (4x4) -> cs[0..15]
  if (lane < 16) {
    int i = lane >> 2, l = lane & 3;
    float s = 0.f;
    #pragma unroll
    for (int t = 0; t < 16; ++t)
      s += As[(16 + i) * LDX + t] * W.lam[t] * As[(16 + l) * LDX + t];
    W.cs[lane] = s;
  }
  __syncthreads();

  // ---- 4x4 log-eig + FC + log_softmax (scalar, lane 0)
  if (lane == 0 && gw < B) {
    float a[4][4], u[4][4];
    #pragma unroll
    for (int i = 0; i < 4; ++i)
      #pragma unroll
      for (int j = 0; j < 4; ++j) {
        a[i][j] = W.cs[i * 4 + j];
        u[i][j] = (i == j) ? 1.f : 0.f;
      }
    for (int sweep = 0; sweep < 12; ++sweep) {
      #pragma unroll
      for (int p = 0; p < 3; ++p)
        #pragma unroll
        for (int q = p + 1; q < 4; ++q) {
          float apq = a[p][q];
          float c = 1.f, s = 0.f;
          if (fabsf(apq) > 1e-30f) {
            float tau = (a[q][q] - a[p][p]) / (2.f * apq);
            float t = copysignf(1.f, tau) / (fabsf(tau) + sqrtf(1.f + tau * tau));
            c = 1.f / sqrtf(1.f + t * t);
            s = t * c;
          }
          #pragma unroll
          for (int j = 0; j < 4; ++j) {
            float xp = a[p][j], xq = a[q][j];
            a[p][j] = c * xp - s * xq;
            a[q][j] = s * xp + c * xq;
          }
          #pragma unroll
          for (int i = 0; i < 4; ++i) {
            float xp = a[i][p], xq = a[i][q];
            a[i][p] = c * xp - s * xq;
            a[i][q] = s * xp + c * xq;
            float up = u[i][p], uq = u[i][q];
            u[i][p] = c * up - s * uq;
            u[i][q] = s * up + c * uq;
          }
        }
    }
    float ll[4];
    #pragma unroll
    for (int t = 0; t < 4; ++t) ll[t] = logf(fmaxf(a[t][t], LOG_EPS));
    float feat[16];
    #pragma unroll
    for (int i = 0; i < 4; ++i)
      #pragma unroll
      for (int j = 0; j < 4; ++j) {
        float s = 0.f;
        #pragma unroll
        for (int t = 0; t < 4; ++t) s += u[i][t] * ll[t] * u[j][t];
        feat[i * 4 + j] = s;
      }
    float l0 = 0.f, l1 = 0.f;
    #pragma unroll
    for (int f = 0; f < 16; ++f) {
      l0 += feat[f] * fcw[f * 2 + 0];
      l1 += feat[f] * fcw[f * 2 + 1];
    }
    float m = fmaxf(l0, l1);
    float lse = m + logf(expf(l0 - m) + expf(l1 - m));
    out[(size_t)b * 2 + 0] = l0 - lse;
    out[(size_t)b * 2 + 1] = l1 - lse;
    float* fo = out + (size_t)B * 2 + (size_t)b * 16;
    #pragma unroll
    for (int f = 0; f < 16; ++f) fo[f] = feat[f];
  }
}

extern "C" void kernel_launch(void* const* d_in, const int* in_sizes, int n_in,
                              void* d_out, int out_size, void* d_ws, size_t ws_size,
                              hipStream_t stream) {
  const float* x   = (const float*)d_in[0];
  const float* w1  = (const float*)d_in[1];
  const float* w2  = (const float*)d_in[2];
  const float* w3  = (const float*)d_in[3];
  const float* fcw = (const float*)d_in[4];
  float* out = (float*)d_out;
  int B = in_sizes[0] / 1024;            // [B,32,32]
  int blocks = (B + WAVES_PER_BLOCK - 1) / WAVES_PER_BLOCK;
  spdnet_fused_kernel<<<blocks, WAVES_PER_BLOCK * 32, 0, stream>>>(
      x, w1, w2, w3, fcw, out, B);
}